// Y_50259707298067
// MI455X (gfx1250) — compile-verified
//
#include <hip/hip_runtime.h>

// Pauli-Y on qubit with bit-stride 2^18 of a 2^24-amplitude complex64 state.
//   lo' = -i * hi   ->  (re, im) = ( hi_im, -hi_re)
//   hi' = +i * lo   ->  (re, im) = (-lo_im,  lo_re)
// Inputs: planar float32 real/imag arrays [N]. Output: interleaved complex64
// viewed as 2N floats.
//
// Pure streaming op: 16 B of obligatory HBM traffic per amplitude, AI ~ 0.
// Optimal kernel = coalesced b128 loads/stores + non-temporal hints so the
// single-pass 256 MB stream doesn't pollute the 192 MB L2. WMMA cannot help:
// it adds no byte savings and only operand-marshalling overhead.

typedef float f4 __attribute__((ext_vector_type(4)));

#define PSTRIDE (1u << 18)   // partner stride (qudit index 5: post = 2^24 / (2^5 * 2))

__global__ __launch_bounds__(256) void pauliY_b18_kernel(
    const float* __restrict__ xr,
    const float* __restrict__ xi,
    float* __restrict__ out)
{
    // t in [0, 2^21): one thread = 4 consecutive low-half amplitudes + 4 partners.
    const unsigned t      = blockIdx.x * blockDim.x + threadIdx.x;
    const unsigned group  = t >> 16;          // p index (0..31), superblock of 2^19 amps
    const unsigned within = t & 0xFFFFu;      // vec4 slot inside the 2^18 low half
    const unsigned n      = (group << 19) + (within << 2);  // flat amp index, bit18 == 0
    const unsigned v      = n >> 2;                         // float4 index (low)
    const unsigned vp     = v + (PSTRIDE >> 2);             // float4 index (partner)

    const f4* __restrict__ xr4  = (const f4*)xr;
    const f4* __restrict__ xi4  = (const f4*)xi;
    f4* __restrict__       out4 = (f4*)out;

    // 4x global_load_b128, non-temporal (read exactly once)
    f4 loR = __builtin_nontemporal_load(xr4 + v);
    f4 loI = __builtin_nontemporal_load(xi4 + v);
    f4 hiR = __builtin_nontemporal_load(xr4 + vp);
    f4 hiI = __builtin_nontemporal_load(xi4 + vp);

    // Interleave + sign flip in registers.
    // low half amps n..n+3: (re, im) = (hiI, -hiR)
    f4 o0 = { hiI.x, -hiR.x, hiI.y, -hiR.y };
    f4 o1 = { hiI.z, -hiR.z, hiI.w, -hiR.w };
    // high half amps n+P..n+P+3: (re, im) = (-loI, loR)
    f4 o2 = { -loI.x, loR.x, -loI.y, loR.y };
    f4 o3 = { -loI.z, loR.z, -loI.w, loR.w };

    // 4x global_store_b128, non-temporal (written once, read back by host)
    __builtin_nontemporal_store(o0, out4 + 2u * v);
    __builtin_nontemporal_store(o1, out4 + 2u * v + 1u);
    __builtin_nontemporal_store(o2, out4 + 2u * vp);
    __builtin_nontemporal_store(o3, out4 + 2u * vp + 1u);
}

extern "C" void kernel_launch(void* const* d_in, const int* in_sizes, int n_in,
                              void* d_out, int out_size, void* d_ws, size_t ws_size,
                              hipStream_t stream)
{
    const float* xr  = (const float*)d_in[0];   // x_real, N float32
    const float* xi  = (const float*)d_in[1];   // x_imag, N float32
    float*       out = (float*)d_out;           // complex64 [N] -> 2N interleaved floats

    const unsigned N       = (unsigned)in_sizes[0];   // 2^24
    const unsigned threads = N / 8u;                  // 4 low amps + 4 high amps per thread
    const unsigned block   = 256u;                    // 8 wave32s
    const unsigned grid    = threads / block;         // 8192

    pauliY_b18_kernel<<<grid, block, 0, stream>>>(xr, xi, out);
}